// YOLOLoss_88742614270491
// MI455X (gfx1250) — compile-verified
//
#include <hip/hip_runtime.h>

// ---------------- constants from the reference ----------------
#define BATCH       128
#define NGT         50
#define NUM_GT      (BATCH * NGT)      // 6400
#define NUM_CLASSES 20
#define INPUT_SZ    640.0f
#define T_SMALL     25600.0f           // (640/4)^2
#define T_MED       102400.0f          // (640/2)^2
#define LAMBDA_BOX  7.5f
#define LAMBDA_CLS  0.5f

#define BLOCK       256
#define NBLK        (NUM_GT / BLOCK)   // 25 blocks, exact

typedef __attribute__((ext_vector_type(2))) float v2f;
typedef __attribute__((ext_vector_type(8))) float v8f;

// -------- kernel 1: one thread per GT, gather + loss terms + block reduce ----
__global__ __launch_bounds__(BLOCK) void yolo_loss_main(
    const float* __restrict__ cls0, const float* __restrict__ box0,
    const float* __restrict__ cls1, const float* __restrict__ box1,
    const float* __restrict__ cls2, const float* __restrict__ box2,
    const float* __restrict__ gt_boxes, const int* __restrict__ gt_labels,
    float* __restrict__ partials)
{
    const int tid = blockIdx.x * BLOCK + threadIdx.x;   // 0..6399, exact coverage

    // ---- GT box in pixel space (xyxy) ----
    const float4 g = ((const float4*)gt_boxes)[tid];    // 16B aligned
    const float x1 = g.x * INPUT_SZ, y1 = g.y * INPUT_SZ;
    const float x2 = g.z * INPUT_SZ, y2 = g.w * INPUT_SZ;
    const float gw = x2 - x1, gh = y2 - y1;
    const float area = gw * gh;

    // ---- scale selection: 0 small, 1 medium, 2 large ----
    const int s = (area < T_SMALL) ? 0 : ((area < T_MED) ? 1 : 2);
    const int G = 80 >> s;                 // 80, 40, 20
    const float stride = (float)(8 << s);  // 8, 16, 32
    const float inv_stride = 1.0f / stride;

    const float cx = (x1 + x2) * 0.5f;
    const float cy = (y1 + y2) * 0.5f;
    int gx = (int)(cx * inv_stride);  gx = min(max(gx, 0), G - 1);
    int gy = (int)(cy * inv_stride);  gy = min(max(gy, 0), G - 1);

    const float* __restrict__ clsp = (s == 0) ? cls0 : ((s == 1) ? cls1 : cls2);
    const float* __restrict__ boxp = (s == 0) ? box0 : ((s == 1) ? box1 : box2);

    const int b = tid / NGT;
    const long cell = ((long)b * G + gy) * G + gx;

    // ---- gather: 20 class probs (5 x b128, 80B stride keeps 16B alignment) ----
    const float4* __restrict__ c4 = (const float4*)(clsp + cell * NUM_CLASSES);
    const float4 pb = ((const float4*)boxp)[cell];
    const int label = gt_labels[tid];

    // ---- BCE over classes: -(1/C) * sum_c [ onehot*logp + (1-onehot)*log1mp ] ----
    float ssum = 0.0f;
#pragma unroll
    for (int q = 0; q < 5; ++q) {
        const float4 p = c4[q];
        const float pv[4] = {p.x, p.y, p.z, p.w};
#pragma unroll
        for (int j = 0; j < 4; ++j) {
            const int c = q * 4 + j;
            const float lp   = fmaxf(logf(pv[j]),           -100.0f);
            const float l1mp = fmaxf(log1pf(-pv[j]),        -100.0f);
            ssum += (c == label) ? lp : l1mp;
        }
    }
    const float cls_term = -ssum * (1.0f / (float)NUM_CLASSES);

    // ---- decode predicted box, IoU with GT ----
    const float px = ((float)gx + pb.x) * stride;
    const float py = ((float)gy + pb.y) * stride;
    const float pw = pb.z * INPUT_SZ;
    const float ph = pb.w * INPUT_SZ;
    const float bx1 = px - pw * 0.5f, by1 = py - ph * 0.5f;
    const float bx2 = px + pw * 0.5f, by2 = py + ph * 0.5f;

    const float ix1 = fmaxf(bx1, x1), iy1 = fmaxf(by1, y1);
    const float ix2 = fminf(bx2, x2), iy2 = fminf(by2, y2);
    const float inter = fmaxf(ix2 - ix1, 0.0f) * fmaxf(iy2 - iy1, 0.0f);
    const float a1 = (bx2 - bx1) * (by2 - by1);
    const float a2 = gw * gh;
    const float iou = inter / (a1 + a2 - inter + 1e-6f);
    float box_term = 1.0f - iou;
    float cls_acc  = cls_term;

    // ---- wave32 tree reduction ----
#pragma unroll
    for (int off = 16; off > 0; off >>= 1) {
        box_term += __shfl_xor(box_term, off, 32);
        cls_acc  += __shfl_xor(cls_acc,  off, 32);
    }

    // ---- cross-wave reduce via LDS (8 waves / block on wave32) ----
    __shared__ float sb[BLOCK / 32];
    __shared__ float sc[BLOCK / 32];
    const int wave = threadIdx.x >> 5;
    const int lane = threadIdx.x & 31;
    if (lane == 0) { sb[wave] = box_term; sc[wave] = cls_acc; }
    __syncthreads();
    if (threadIdx.x == 0) {
        float tb = 0.0f, tc = 0.0f;
#pragma unroll
        for (int i = 0; i < BLOCK / 32; ++i) { tb += sb[i]; tc += sc[i]; }
        partials[2 * blockIdx.x + 0] = tb;
        partials[2 * blockIdx.x + 1] = tc;
    }
}

// -------- kernel 2: one wave; exact f32 WMMA row-sum reduction of partials ----
// A (16x4 f32, documented layout): rows 0-7 hold the 25 box partials at slots
// j = k*8 + row; rows 8-15 hold the 25 cls partials. B = all-ones (layout-
// independent). D[m][n] = sum_k A[m][k] -> lane 0's 8 accumulators are rows
// 0..7 (box total), lane 16's are rows 8..15 (cls total).
__global__ __launch_bounds__(32) void yolo_loss_finalize(
    const float* __restrict__ partials, float* __restrict__ out)
{
    const int L = threadIdx.x;          // 0..31, EXEC all ones
    const int m  = L & 15;              // A-matrix row for this lane
    const int k0 = (L < 16) ? 0 : 2;    // VGPR0 holds K=0 (lanes 0-15) / K=2 (16-31)

    auto fetch = [&](int mm, int kk) -> float {
        const int row   = mm & 7;
        const int which = (mm >= 8) ? 1 : 0;   // 0 = box partial, 1 = cls partial
        const int j     = kk * 8 + row;
        return (j < NBLK) ? partials[2 * j + which] : 0.0f;
    };

    v2f a;  a.x = fetch(m, k0);  a.y = fetch(m, k0 + 1);
    v2f bones;  bones.x = 1.0f;  bones.y = 1.0f;
    v8f c = {};
    // 8 args: (neg_a, A, neg_b, B, c_mod, C, reuse_a, reuse_b)
    c = __builtin_amdgcn_wmma_f32_16x16x4_f32(
        false, a, false, bones, (short)0, c, false, false);

    float rowsum = c[0] + c[1] + c[2] + c[3] + c[4] + c[5] + c[6] + c[7];
    const float tot_box = __shfl(rowsum, 0, 32);   // D rows 0..7 @ N=0
    const float tot_cls = __shfl(rowsum, 16, 32);  // D rows 8..15 @ N=0

    if (L == 0) {
        const float inv_gt   = 1.0f / (float)NUM_GT;
        const float box_loss = tot_box * inv_gt;
        const float cls_loss = tot_cls * inv_gt;
        out[0] = LAMBDA_BOX * box_loss + LAMBDA_CLS * cls_loss;  // total
        out[1] = box_loss;
        out[2] = cls_loss;
        out[3] = 0.0f;                                           // obj_loss
    }
}

extern "C" void kernel_launch(void* const* d_in, const int* in_sizes, int n_in,
                              void* d_out, int out_size, void* d_ws, size_t ws_size,
                              hipStream_t stream) {
    const float* cls0 = (const float*)d_in[0];
    const float* box0 = (const float*)d_in[1];
    const float* cls1 = (const float*)d_in[2];
    const float* box1 = (const float*)d_in[3];
    const float* cls2 = (const float*)d_in[4];
    const float* box2 = (const float*)d_in[5];
    const float* gtb  = (const float*)d_in[6];
    const int*   gtl  = (const int*)  d_in[7];
    float* out      = (float*)d_out;
    float* partials = (float*)d_ws;     // 2 * NBLK floats

    yolo_loss_main<<<NBLK, BLOCK, 0, stream>>>(cls0, box0, cls1, box1,
                                               cls2, box2, gtb, gtl, partials);
    yolo_loss_finalize<<<1, 32, 0, stream>>>(partials, out);
}